// LETTERQuantizer_4140348473630
// MI455X (gfx1250) — compile-verified
//
#include <hip/hip_runtime.h>

#define BN 65536
#define DN 64
#define KN 1024
#define SK_EPS 0.05f
#define SK_ITERS 100
#define SINK_BLOCKS 256   // persistent kernel grid: 256 blocks x 8 waves, 32 rows/wave

typedef float v2f __attribute__((ext_vector_type(2)));
typedef float v8f __attribute__((ext_vector_type(8)));

__device__ __forceinline__ unsigned fenc(float f) {
  unsigned u = __float_as_uint(f);
  return (u & 0x80000000u) ? ~u : (u | 0x80000000u);
}
__device__ __forceinline__ float fdec(unsigned e) {
  unsigned u = (e & 0x80000000u) ? (e ^ 0x80000000u) : ~e;
  return __uint_as_float(u);
}
__device__ __forceinline__ unsigned short f2bf(float f) {
  unsigned u = __float_as_uint(f);
  return (unsigned short)((u + 0x7FFFu + ((u >> 16) & 1u)) >> 16);
}
__device__ __forceinline__ v2f unpk(unsigned dw) {
  v2f r;
  r.x = __uint_as_float(dw << 16);
  r.y = __uint_as_float(dw & 0xFFFF0000u);
  return r;
}

// -------- init: c buffers (triple), minmax sentinels, barrier cnt, outputs --
__global__ void k_init(float* c0, float* c1, float* c2, unsigned* mm,
                       unsigned* cnt, float* loss, float* zslot) {
  int t = threadIdx.x;
  if (t < KN) {
    c0[t] = 0.0f;                 // accumulated by iteration 0
    c1[t] = 0.0f;                 // zeroed again by iteration 0's rotation
    c2[t] = 1.0f / 1024.0f;       // read by iteration 0 -> v = 1 (reference init)
  }
  if (t == 0) { mm[0] = 0xFFFFFFFFu; mm[1] = 0u; *cnt = 0u; *loss = 0.0f; *zslot = 0.0f; }
}

// -------- squared row norms of x (B) and codebook (K) ----------------------
__global__ void k_norms(const float* __restrict__ x, const float* __restrict__ cb,
                        float* __restrict__ xn, float* __restrict__ cn) {
  int tid = blockIdx.x * blockDim.x + threadIdx.x;
  if (tid < KN) {
    const float4* p = (const float4*)(cb + (size_t)tid * DN);
    float s = 0.f;
#pragma unroll
    for (int i = 0; i < DN / 4; ++i) { float4 q = p[i]; s += q.x*q.x + q.y*q.y + q.z*q.z + q.w*q.w; }
    cn[tid] = s;
  } else {
    int i = tid - KN;
    if (i < BN) {
      const float4* p = (const float4*)(x + (size_t)i * DN);
      float s = 0.f;
#pragma unroll
      for (int k = 0; k < DN / 4; ++k) { float4 q = p[k]; s += q.x*q.x + q.y*q.y + q.z*q.z + q.w*q.w; }
      xn[i] = s;
    }
  }
}

// -------- pass 1: WMMA GEMM, track global min/max of d (d discarded) -------
__global__ __launch_bounds__(256) void k_minmax(
    const float* __restrict__ x, const float* __restrict__ cb,
    const float* __restrict__ xn, const float* __restrict__ cn,
    unsigned* __restrict__ mm) {
  int lane = threadIdx.x & 31;
  int gw = blockIdx.x * 8 + (threadIdx.x >> 5);
  int row0 = gw * 16;
  int m15 = lane & 15, hi = lane >> 4;

  v2f a[16];
  const float* xrow = x + (size_t)(row0 + m15) * DN + 2 * hi;
#pragma unroll
  for (int k = 0; k < 16; ++k) a[k] = *(const v2f*)(xrow + 4 * k);
  float xr[8];
#pragma unroll
  for (int r = 0; r < 8; ++r) xr[r] = xn[row0 + r + 8 * hi];

  float lmin = 3.4e38f, lmax = -3.4e38f;
  for (int t = 0; t < KN / 16; ++t) {
    int col0 = t * 16;
    const float* brow = cb + (size_t)(col0 + m15) * DN + 2 * hi;
    v8f acc = {0.f, 0.f, 0.f, 0.f, 0.f, 0.f, 0.f, 0.f};
#pragma unroll
    for (int k = 0; k < 16; ++k) {
      v2f b = *(const v2f*)(brow + 4 * k);
      acc = __builtin_amdgcn_wmma_f32_16x16x4_f32(false, a[k], false, b, (short)0, acc, false, false);
    }
    float cnn = cn[col0 + m15];
#pragma unroll
    for (int r = 0; r < 8; ++r) {
      float d = xr[r] + cnn - 2.0f * acc[r];
      lmin = fminf(lmin, d);
      lmax = fmaxf(lmax, d);
    }
  }
#pragma unroll
  for (int m = 16; m >= 1; m >>= 1) {
    lmin = fminf(lmin, __shfl_xor(lmin, m, 32));
    lmax = fmaxf(lmax, __shfl_xor(lmax, m, 32));
  }
  if (lane == 0) { atomicMin(&mm[0], fenc(lmin)); atomicMax(&mm[1], fenc(lmax)); }
}

// -------- decode min/max -> mn, 1/(amp*eps)*log2(e), amp*eps ---------------
__global__ void k_decode(const unsigned* mm, float* par) {
  float mn = fdec(mm[0]), mx = fdec(mm[1]);
  float mid = 0.5f * (mx + mn);
  float amp = mx - mid + 1e-5f;
  par[0] = mn;
  par[1] = 1.4426950408889634f / (amp * SK_EPS);
  par[2] = amp * SK_EPS;
}

// -------- pass 2: recompute d via WMMA, store E = exp(...) as bf16 ---------
__global__ __launch_bounds__(256) void k_expstore(
    const float* __restrict__ x, const float* __restrict__ cb,
    const float* __restrict__ xn, const float* __restrict__ cn,
    const float* __restrict__ par, unsigned short* __restrict__ E) {
  int lane = threadIdx.x & 31;
  int gw = blockIdx.x * 8 + (threadIdx.x >> 5);
  int row0 = gw * 16;
  int m15 = lane & 15, hi = lane >> 4;
  float mn = par[0], c1 = par[1];

  v2f a[16];
  const float* xrow = x + (size_t)(row0 + m15) * DN + 2 * hi;
#pragma unroll
  for (int k = 0; k < 16; ++k) a[k] = *(const v2f*)(xrow + 4 * k);
  float xr[8];
#pragma unroll
  for (int r = 0; r < 8; ++r) xr[r] = xn[row0 + r + 8 * hi];

  for (int t = 0; t < KN / 16; ++t) {
    int col0 = t * 16;
    const float* brow = cb + (size_t)(col0 + m15) * DN + 2 * hi;
    v8f acc = {0.f, 0.f, 0.f, 0.f, 0.f, 0.f, 0.f, 0.f};
#pragma unroll
    for (int k = 0; k < 16; ++k) {
      v2f b = *(const v2f*)(brow + 4 * k);
      acc = __builtin_amdgcn_wmma_f32_16x16x4_f32(false, a[k], false, b, (short)0, acc, false, false);
    }
    float cnn = cn[col0 + m15];
#pragma unroll
    for (int r = 0; r < 8; ++r) {
      float d = xr[r] + cnn - 2.0f * acc[r];
      float e = exp2f(-(d - mn) * c1);
      E[(size_t)(row0 + r + 8 * hi) * KN + col0 + m15] = f2bf(e);
    }
  }
}

// -------- persistent fused Sinkhorn: all 100 iterations in one kernel ------
// per iter: v_j = rcp(K*cprev_j); r_i = E_i . v; u_i = rcp(B*r_i);
// c_j += E_ij*u_i  (regs -> LDS block partial -> 1024 global atomics/block)
__global__ __launch_bounds__(256) void k_sinkhorn(
    const unsigned short* __restrict__ E,
    float* __restrict__ c0, float* __restrict__ c1, float* __restrict__ c2,
    unsigned* __restrict__ cnt) {
  __shared__ float part[8][KN];   // 32 KB block partial column sums

  int lane = threadIdx.x & 31;
  int wv = threadIdx.x >> 5;
  int gw = blockIdx.x * 8 + wv;
  int row0 = gw * 32;
  const unsigned short* Ew = E + (size_t)row0 * KN;

  float* bufs[3] = {c0, c1, c2};

  for (int it = 0; it < SK_ITERS; ++it) {
    float* cacc = bufs[it % 3];
    const float* cprev = bufs[(it + 2) % 3];
    float* czero = bufs[(it + 1) % 3];

    // zero the buffer the NEXT iteration accumulates into (race-free: nobody
    // reads or writes it during this iteration)
    if (threadIdx.x < 4) czero[blockIdx.x * 4 + threadIdx.x] = 0.f;

    // per-lane v for its 32 fixed columns (coherent agent-scope loads)
    v2f vv[16];
#pragma unroll
    for (int i = 0; i < 16; ++i) {
      int jb = ((i >> 2) * 32 + lane) * 8 + 2 * (i & 3);
      float ca = __hip_atomic_load(&cprev[jb],     __ATOMIC_RELAXED, __HIP_MEMORY_SCOPE_AGENT);
      float cb2 = __hip_atomic_load(&cprev[jb + 1], __ATOMIC_RELAXED, __HIP_MEMORY_SCOPE_AGENT);
      vv[i].x = __builtin_amdgcn_rcpf(1024.0f * ca + 1e-30f);
      vv[i].y = __builtin_amdgcn_rcpf(1024.0f * cb2 + 1e-30f);
    }

    v2f acc[16];
#pragma unroll
    for (int i = 0; i < 16; ++i) { acc[i].x = 0.f; acc[i].y = 0.f; }

    for (int r = 0; r < 32; ++r) {
      const uint4* p = (const uint4*)(Ew + (size_t)r * KN);
      __builtin_prefetch((const char*)(Ew + (size_t)((r + 4) & 31) * KN) + lane * 64, 0, 0);
      uint4 q0 = p[lane], q1 = p[32 + lane], q2 = p[64 + lane], q3 = p[96 + lane];
      v2f ef[16];
      ef[0] = unpk(q0.x);  ef[1] = unpk(q0.y);  ef[2] = unpk(q0.z);  ef[3] = unpk(q0.w);
      ef[4] = unpk(q1.x);  ef[5] = unpk(q1.y);  ef[6] = unpk(q1.z);  ef[7] = unpk(q1.w);
      ef[8] = unpk(q2.x);  ef[9] = unpk(q2.y);  ef[10] = unpk(q2.z); ef[11] = unpk(q2.w);
      ef[12] = unpk(q3.x); ef[13] = unpk(q3.y); ef[14] = unpk(q3.z); ef[15] = unpk(q3.w);

      v2f sd = ef[0] * vv[0];
#pragma unroll
      for (int i = 1; i < 16; ++i) sd = ef[i] * vv[i] + sd;   // v_pk_fma_f32
      float s = sd.x + sd.y;
#pragma unroll
      for (int m = 16; m >= 1; m >>= 1) s += __shfl_xor(s, m, 32);
      float u = __builtin_amdgcn_rcpf(65536.0f * s + 1e-30f);
      v2f u2; u2.x = u; u2.y = u;
#pragma unroll
      for (int i = 0; i < 16; ++i) acc[i] = ef[i] * u2 + acc[i];
    }

    // wave regs -> LDS partials
#pragma unroll
    for (int i = 0; i < 16; ++i) {
      int jb = ((i >> 2) * 32 + lane) * 8 + 2 * (i & 3);
      part[wv][jb] = acc[i].x;
      part[wv][jb + 1] = acc[i].y;
    }
    __syncthreads();
    // reduce 8 waves, 4 columns per thread, one global atomic per column
#pragma unroll
    for (int cix = 0; cix < 4; ++cix) {
      int col = threadIdx.x * 4 + cix;
      float s = part[0][col];
#pragma unroll
      for (int wi = 1; wi < 8; ++wi) s += part[wi][col];
      atomicAdd(&cacc[col], s);
    }

    // device-wide barrier (monotonic counter, no reset)
    if (it != SK_ITERS - 1) {
      __syncthreads();
      if (threadIdx.x == 0) {
        __threadfence();
        __hip_atomic_fetch_add(cnt, 1u, __ATOMIC_ACQ_REL, __HIP_MEMORY_SCOPE_AGENT);
        unsigned target = (unsigned)SINK_BLOCKS * (unsigned)(it + 1);
        while (__hip_atomic_load(cnt, __ATOMIC_ACQUIRE, __HIP_MEMORY_SCOPE_AGENT) < target)
          __builtin_amdgcn_s_sleep(1);
        __threadfence();
      }
      __syncthreads();
    }
  }
}

// -------- w_j = amp*eps*ln(v_j) from final column sums ---------------------
__global__ void k_prepw(const float* cfin, const float* par, float* w) {
  int t = threadIdx.x;
  w[t] = -par[2] * logf(1024.0f * cfin[t] + 1e-30f);
}

// -------- final: exact-fp32 WMMA argmin + gather + STE out + loss ----------
__global__ __launch_bounds__(256) void k_final(
    const float* __restrict__ x, const float* __restrict__ cb,
    const float* __restrict__ xn, const float* __restrict__ cn,
    const float* __restrict__ w, float* __restrict__ xq,
    int* __restrict__ idx, float* __restrict__ loss) {
  int lane = threadIdx.x & 31;
  int gw = blockIdx.x * 8 + (threadIdx.x >> 5);
  int row0 = gw * 16;
  int m15 = lane & 15, hi = lane >> 4;

  v2f a[16];
  const float* xrow = x + (size_t)(row0 + m15) * DN + 2 * hi;
#pragma unroll
  for (int k = 0; k < 16; ++k) a[k] = *(const v2f*)(xrow + 4 * k);
  float xr[8];
#pragma unroll
  for (int r = 0; r < 8; ++r) xr[r] = xn[row0 + r + 8 * hi];

  float bs[8]; int bj[8];
#pragma unroll
  for (int r = 0; r < 8; ++r) { bs[r] = 3.4e38f; bj[r] = 0; }

  for (int t = 0; t < KN / 16; ++t) {
    int col0 = t * 16;
    const float* brow = cb + (size_t)(col0 + m15) * DN + 2 * hi;
    v8f acc = {0.f, 0.f, 0.f, 0.f, 0.f, 0.f, 0.f, 0.f};
#pragma unroll
    for (int k = 0; k < 16; ++k) {
      v2f b = *(const v2f*)(brow + 4 * k);
      acc = __builtin_amdgcn_wmma_f32_16x16x4_f32(false, a[k], false, b, (short)0, acc, false, false);
    }
    float cw = cn[col0 + m15] - w[col0 + m15];
#pragma unroll
    for (int r = 0; r < 8; ++r) {
      float s = xr[r] + cw - 2.0f * acc[r];
      if (s < bs[r]) { bs[r] = s; bj[r] = col0 + m15; }
    }
  }
  // reduce per half-wave (16 lanes share the same 8 rows), first-index tiebreak
#pragma unroll
  for (int r = 0; r < 8; ++r) {
#pragma unroll
    for (int m = 1; m <= 8; m <<= 1) {
      float os = __shfl_xor(bs[r], m, 32);
      int oj = __shfl_xor(bj[r], m, 32);
      if (os < bs[r] || (os == bs[r] && oj < bj[r])) { bs[r] = os; bj[r] = oj; }
    }
    if (m15 == r) idx[row0 + r + 8 * hi] = bj[r];
  }
  // x_q gather + loss; x_q_ste == x_q numerically; both losses equal -> 1.25*MSE
  float lsum = 0.f;
#pragma unroll
  for (int m = 0; m < 16; ++m) {
    int j = __shfl(bj[m & 7], (m >> 3) * 16, 32);
    int row = row0 + m;
    v2f cv = *(const v2f*)(cb + (size_t)j * DN + 2 * lane);
    v2f xv = *(const v2f*)(x + (size_t)row * DN + 2 * lane);
    *(v2f*)(xq + (size_t)row * DN + 2 * lane) = cv;
    float dx = cv.x - xv.x, dy = cv.y - xv.y;
    lsum = fmaf(dx, dx, fmaf(dy, dy, lsum));
  }
#pragma unroll
  for (int m = 16; m >= 1; m >>= 1) lsum += __shfl_xor(lsum, m, 32);
  if (lane == 0) atomicAdd(loss, lsum * (1.25f / (float)(BN * DN)));
}

extern "C" void kernel_launch(void* const* d_in, const int* in_sizes, int n_in,
                              void* d_out, int out_size, void* d_ws, size_t ws_size,
                              hipStream_t stream) {
  (void)in_sizes; (void)n_in; (void)out_size; (void)ws_size;
  const float* x  = (const float*)d_in[0];
  const float* cb = (const float*)d_in[1];

  // workspace: E (bf16, 128MB) | xn | cn | c0 | c1 | c2 | w | mm | par | cnt
  unsigned short* E = (unsigned short*)d_ws;
  float* xn = (float*)((char*)d_ws + (size_t)BN * KN * 2);
  float* cn = xn + BN;
  float* c0 = cn + KN;
  float* c1 = c0 + KN;
  float* c2 = c1 + KN;
  float* w  = c2 + KN;
  unsigned* mm  = (unsigned*)(w + KN);
  float* par    = (float*)(mm + 2);
  unsigned* cnt = (unsigned*)(par + 3);

  // output: x_q_ste [B*D] | loss [1] | indices(int32) [B] | 0.0 [1]
  float* out   = (float*)d_out;
  float* xq    = out;
  float* loss  = out + (size_t)BN * DN;
  int*   idx   = (int*)(loss + 1);
  float* zslot = (float*)(idx + BN);

  k_init<<<1, 1024, 0, stream>>>(c0, c1, c2, mm, cnt, loss, zslot);
  k_norms<<<(BN + KN + 255) / 256, 256, 0, stream>>>(x, cb, xn, cn);
  k_minmax<<<BN / 128, 256, 0, stream>>>(x, cb, xn, cn, mm);
  k_decode<<<1, 1, 0, stream>>>(mm, par);
  k_expstore<<<BN / 128, 256, 0, stream>>>(x, cb, xn, cn, par, E);
  k_sinkhorn<<<SINK_BLOCKS, 256, 0, stream>>>(E, c0, c1, c2, cnt);
  // final column sums land in bufs[(SK_ITERS-1) % 3] == c0 for 100 iters
  k_prepw<<<1, KN, 0, stream>>>(c0, par, w);
  k_final<<<BN / 128, 256, 0, stream>>>(x, cb, xn, cn, w, xq, idx, loss);
}